// Attention_22617297780737
// MI455X (gfx1250) — compile-verified
//
#include <hip/hip_runtime.h>
#include <math.h>

#define SLEN 8192
#define HDIM 2048
#define W2H  4096
#define NPART 16

typedef float v2f __attribute__((ext_vector_type(2)));
typedef float v8f __attribute__((ext_vector_type(8)));

// ---------------------------------------------------------------------------
// Phase 1: u[j] = sum_o w[o] * W_att[o, H + j]   (only the W_e half matters;
// hidden / b_att / W_h contribute a constant that softmax cancels).
// Grid: (H/256, NPART). Coalesced along j; w[o] is a uniform scalar load.
// ---------------------------------------------------------------------------
__global__ void u_partial_kernel(const float* __restrict__ W_att,
                                 const float* __restrict__ w,
                                 float* __restrict__ partials) {
    const int j  = blockIdx.x * blockDim.x + threadIdx.x;   // 0..H-1
    const int p  = blockIdx.y;                              // 0..NPART-1
    const int o0 = p * (HDIM / NPART);
    const float* base = W_att + (size_t)o0 * W2H + HDIM + j;
    float acc = 0.0f;
#pragma unroll 4
    for (int o = 0; o < HDIM / NPART; ++o) {
        acc = fmaf(w[o0 + o], base[(size_t)o * W2H], acc);
    }
    partials[p * HDIM + j] = acc;
}

__global__ void u_reduce_kernel(const float* __restrict__ partials,
                                float* __restrict__ u) {
    const int j = blockIdx.x * blockDim.x + threadIdx.x;
    float acc = 0.0f;
#pragma unroll
    for (int p = 0; p < NPART; ++p) acc += partials[p * HDIM + j];
    u[j] = acc;
}

// ---------------------------------------------------------------------------
// Phase 2: scores[s] = enc[s] . u   via V_WMMA_F32_16X16X4_F32.
// One wave per 16-row tile of enc. Per step k (K=4):
//   A (16x4 f32): lane m (0..15)  -> enc[row_m][k+0], enc[row_m][k+1]
//                 lane m+16       -> enc[row_m][k+2], enc[row_m][k+3]
//     -> a single 8-byte load per lane: enc + row*H + k + (lane>=16 ? 2 : 0)
//   B (4x16 f32): u[k..k+3] broadcast across all 16 columns, so
//     lane<16: {u[k], u[k+1]}  lane>=16: {u[k+2], u[k+3]}  (same addr pattern)
// Every column of the accumulator then holds the same running dot product.
// ---------------------------------------------------------------------------
__global__ void scores_wmma_kernel(const float* __restrict__ enc,
                                   const float* __restrict__ u,
                                   float* __restrict__ scores) {
    const int lane  = threadIdx.x & 31;
    const int wave  = threadIdx.x >> 5;
    const int tile  = blockIdx.x * (blockDim.x >> 5) + wave;
    const int row   = tile * 16 + (lane & 15);
    const int khalf = (lane >> 4) << 1;          // 0 for lanes 0-15, 2 for 16-31

    const float* arow = enc + (size_t)row * HDIM + khalf;
    const float* ucol = u + khalf;

    v8f acc = {};
#pragma unroll 8
    for (int k = 0; k < HDIM; k += 4) {
        v2f a = *(const v2f*)(arow + k);         // 8B aligned (k+khalf even)
        v2f b = *(const v2f*)(ucol + k);
        acc = __builtin_amdgcn_wmma_f32_16x16x4_f32(
            /*neg_a=*/false, a, /*neg_b=*/false, b,
            /*c_mod=*/(short)0, acc, /*reuse_a=*/false, /*reuse_b=*/false);
    }

    // D layout: VGPR r -> (M=r, N=lane) for lanes 0-15, (M=8+r) for lanes 16-31.
    // Column 0 lives in lane 0 (rows 0-7) and lane 16 (rows 8-15).
    if (lane == 0) {
#pragma unroll
        for (int r = 0; r < 8; ++r) scores[tile * 16 + r] = acc[r];
    }
    if (lane == 16) {
#pragma unroll
        for (int r = 0; r < 8; ++r) scores[tile * 16 + 8 + r] = acc[r];
    }
}

// ---------------------------------------------------------------------------
// Phase 3: softmax over scores[SLEN]. Single 1024-thread block, fixed-order
// tree reductions (deterministic bit-exact across replays).
// ---------------------------------------------------------------------------
__global__ void softmax_kernel(const float* __restrict__ scores,
                               float* __restrict__ out) {
    __shared__ float red[1024];
    const int t = threadIdx.x;

    float m = -INFINITY;
    for (int i = t; i < SLEN; i += 1024) m = fmaxf(m, scores[i]);
    red[t] = m;
    __syncthreads();
    for (int s = 512; s > 0; s >>= 1) {
        if (t < s) red[t] = fmaxf(red[t], red[t + s]);
        __syncthreads();
    }
    m = red[0];
    __syncthreads();

    float sum = 0.0f;
    for (int i = t; i < SLEN; i += 1024) sum += expf(scores[i] - m);
    red[t] = sum;
    __syncthreads();
    for (int s = 512; s > 0; s >>= 1) {
        if (t < s) red[t] += red[t + s];
        __syncthreads();
    }
    const float inv = 1.0f / red[0];

    for (int i = t; i < SLEN; i += 1024) out[i] = expf(scores[i] - m) * inv;
}

// ---------------------------------------------------------------------------
extern "C" void kernel_launch(void* const* d_in, const int* in_sizes, int n_in,
                              void* d_out, int out_size, void* d_ws, size_t ws_size,
                              hipStream_t stream) {
    const float* enc   = (const float*)d_in[0];   // (S,1,H) fp32
    // d_in[1] = hidden, d_in[3] = b_att : mathematically cancelled by softmax
    const float* W_att = (const float*)d_in[2];   // (H, 2H) fp32
    const float* w     = (const float*)d_in[4];   // (1, H)  fp32
    float* out = (float*)d_out;                   // (1,1,S) fp32

    float* partials = (float*)d_ws;               // NPART*H floats
    float* u        = partials + NPART * HDIM;    // H floats
    float* scores   = u + HDIM;                   // S floats

    u_partial_kernel<<<dim3(HDIM / 256, NPART), 256, 0, stream>>>(W_att, w, partials);
    u_reduce_kernel<<<HDIM / 256, 256, 0, stream>>>(partials, u);
    // 512 tiles of 16 rows, 8 waves (256 threads) per block -> 64 blocks
    scores_wmma_kernel<<<SLEN / 16 / 8, 256, 0, stream>>>(enc, u, scores);
    softmax_kernel<<<1, 1024, 0, stream>>>(scores, out);
}